// TSNELoss_34428457845460
// MI455X (gfx1250) — compile-verified
//
#include <hip/hip_runtime.h>
#include <hip/hip_bf16.h>
#include <math.h>

// ---------------- problem constants ----------------
#define NPTS   8192
#define DIM    128
#define KNN    91            // reference keeps argsort ranks 1..91
#define EPS_F  2.2204460492503131e-16f
#define LSTR   132           // LDS row stride in floats: 16B-aligned rows, conflict-free WMMA fetch

typedef float v2f __attribute__((ext_vector_type(2)));
typedef float v8f __attribute__((ext_vector_type(8)));

// ---------------- K1: row squared norms ----------------
__global__ __launch_bounds__(256) void sq_kernel(const float* __restrict__ x,
                                                 float* __restrict__ sq) {
  const int lane = threadIdx.x & 31;
  const int wv   = threadIdx.x >> 5;
  const int row  = blockIdx.x * 8 + wv;              // grid = 1024 -> rows 0..8191
  float4 v = *(const float4*)(x + (size_t)row * DIM + lane * 4);
  float s = v.x * v.x + v.y * v.y + v.z * v.z + v.w * v.w;
  #pragma unroll
  for (int off = 16; off > 0; off >>= 1) s += __shfl_down(s, off, 32);
  if (lane == 0) sq[row] = s;
}

// ---------------- K2: fused WMMA distance GEMM + top-91 selection + Pn ----------------
// One block = 16 rows. 8 waves each own one 16x16 WMMA tile -> 128 columns/iter, 64 iters.
__global__ __launch_bounds__(256) void knn_kernel(const float* __restrict__ x,
                                                  const float* __restrict__ sq,
                                                  const float* __restrict__ beta,
                                                  int*   __restrict__ knn_idx,
                                                  float* __restrict__ knn_p,
                                                  float* __restrict__ rowPart) {
  __shared__ float ldsA[16 * LSTR];                  //  8.4 KB  A panel (16 x 128)
  __shared__ float ldsB[128 * LSTR];                 // 67.6 KB  B panel (128 x 128)
  __shared__ float ldsDist[16 * LSTR];               //  8.4 KB  distance tile
  __shared__ float sqm[16];
  __shared__ float sqj[128];
  __shared__ float listD[16][KNN];                   // per-row top-91 distances
  __shared__ int   listI[16][KNN];                   // per-row top-91 indices
  __shared__ float rowS[16];

  const int tid  = threadIdx.x;
  const int lane = tid & 31;
  const int wv   = tid >> 5;                         // wave 0..7 (wave32)
  const int la   = lane & 15;                        // N / M-half index
  const int hb   = lane >> 4;                        // K-half select
  const int m0   = blockIdx.x * 16;

  // stage A panel (rows m0..m0+15, contiguous in x) with padded stride
  for (int idx = tid; idx < 16 * 32; idx += 256) {
    const int r = idx >> 5, c4 = idx & 31;
    float4 v = *(const float4*)(x + (size_t)(m0 + r) * DIM + c4 * 4);
    *(float4*)(&ldsA[r * LSTR + c4 * 4]) = v;
  }
  if (tid < 16) sqm[tid] = sq[m0 + tid];

  int   cnt  = 0;                                    // selection state (threads 0..15 only)
  float maxv = -1e30f;
  int   maxp = 0;
  __syncthreads();

  for (int j0 = 0; j0 < NPTS; j0 += 128) {
    // stage B panel (rows j0..j0+127 of x, contiguous)
    for (int idx = tid; idx < 128 * 32; idx += 256) {
      const int r = idx >> 5, c4 = idx & 31;
      float4 v = *(const float4*)(x + (size_t)(j0 + r) * DIM + c4 * 4);
      *(float4*)(&ldsB[r * LSTR + c4 * 4]) = v;
    }
    if (tid < 128) sqj[tid] = sq[j0 + tid];
    if (j0 + 128 < NPTS)                              // hint next panel -> global_prefetch_b8
      __builtin_prefetch(x + (size_t)(j0 + 128) * DIM + tid * 64, 0, 0);
    __syncthreads();

    // ---- fp32 WMMA: C[m][n] = sum_k A[m][k] * B[k][n], K accumulated 4 at a time ----
    v8f acc = {0.f, 0.f, 0.f, 0.f, 0.f, 0.f, 0.f, 0.f};
    #pragma unroll
    for (int kk = 0; kk < DIM; kk += 4) {
      // A 16x4 fp32: lane holds row la, K-pair kk+2*hb .. +1
      v2f a = *(const v2f*)(&ldsA[la * LSTR + kk + 2 * hb]);
      // B 4x16 fp32: lane holds col la of x-row (j0+16*wv+la), same K-pair
      v2f b = *(const v2f*)(&ldsB[(wv * 16 + la) * LSTR + kk + 2 * hb]);
      acc = __builtin_amdgcn_wmma_f32_16x16x4_f32(false, a, false, b,
                                                  (short)0, acc, false, false);
    }

    // distances:  d = |xi|^2 + |xj|^2 - 2 xi.xj  -> LDS tile
    #pragma unroll
    for (int v = 0; v < 8; ++v) {
      const int M = v + 8 * hb;                      // C layout: VGPR v, lane half -> row
      const float dv = sqm[M] + sqj[wv * 16 + la] - 2.0f * acc[v];
      ldsDist[M * LSTR + wv * 16 + la] = dv;
    }
    __syncthreads();

    // ---- per-row streaming top-91 (smallest) selection, one thread per row ----
    if (tid < 16) {
      const int i = m0 + tid;
      for (int c = 0; c < 128; ++c) {
        const int j = j0 + c;
        if (j == i) continue;                        // exclude self (argsort rank 0)
        const float dc = ldsDist[tid * LSTR + c];
        if (cnt < KNN) {
          listD[tid][cnt] = dc; listI[tid][cnt] = j;
          if (dc > maxv) { maxv = dc; maxp = cnt; }
          ++cnt;
        } else if (dc < maxv) {
          listD[tid][maxp] = dc; listI[tid][maxp] = j;
          float mv = listD[tid][0]; int mp = 0;
          for (int q = 1; q < KNN; ++q) {
            const float lv = listD[tid][q];
            if (lv > mv) { mv = lv; mp = q; }
          }
          maxv = mv; maxp = mp;
        }
      }
    }
    __syncthreads();                                 // protect ldsB/ldsDist reuse
  }

  // ---- epilogue: Pn = exp(-d*beta) / sum, emit sparse row + row-sum partial ----
  if (tid < 16) {
    const int i = m0 + tid;
    float b = beta[i];
    if (b == 0.0f) b = EPS_F;
    float s = 0.f;
    for (int q = 0; q < KNN; ++q) {
      const float nm = __expf(-listD[tid][q] * b);
      listD[tid][q] = nm;
      s += nm;
    }
    float sp = 0.f;
    for (int q = 0; q < KNN; ++q) {
      const float p = listD[tid][q] / s;
      knn_p  [(size_t)i * KNN + q] = p;
      knn_idx[(size_t)i * KNN + q] = listI[tid][q];
      sp += p;
    }
    rowS[tid] = 2.0f * sp;                           // Pf + Pf^T total mass contribution
  }
  __syncthreads();
  if (tid == 0) {
    float t = 0.f;
    for (int q = 0; q < 16; ++q) t += rowS[q];
    rowPart[blockIdx.x] = t;
  }
}

// ---------------- K3: dense y-space sums: denom = sum 1/(1+d), logSum = sum log1p(d) ----------------
__global__ __launch_bounds__(256) void qsum_kernel(const float* __restrict__ y,
                                                   float* __restrict__ qnumPart,
                                                   float* __restrict__ qlogPart) {
  __shared__ float ys[2 * NPTS];                     // 64 KB: all of y in LDS
  __shared__ float red[256];
  const int tid = threadIdx.x;
  for (int idx = tid * 4; idx < 2 * NPTS; idx += 1024)
    *(float4*)(&ys[idx]) = *(const float4*)(y + idx);
  __syncthreads();

  const int i = blockIdx.x * 256 + tid;              // grid = 32 -> one row per thread
  const float yi0 = ys[2 * i], yi1 = ys[2 * i + 1];
  float sn = 0.f, sl = 0.f;
  for (int j = 0; j < NPTS; ++j) {
    if (j == i) continue;
    const float dx = yi0 - ys[2 * j];
    const float dy = yi1 - ys[2 * j + 1];
    const float d  = dx * dx + dy * dy;
    sn += 1.0f / (1.0f + d);
    sl += log1pf(d);
  }
  red[tid] = sn; __syncthreads();
  for (int off = 128; off > 0; off >>= 1) { if (tid < off) red[tid] += red[tid + off]; __syncthreads(); }
  if (tid == 0) qnumPart[blockIdx.x] = red[0];
  __syncthreads();
  red[tid] = sl; __syncthreads();
  for (int off = 128; off > 0; off >>= 1) { if (tid < off) red[tid] += red[tid + off]; __syncthreads(); }
  if (tid == 0) qlogPart[blockIdx.x] = red[0];
}

// ---------------- K3b: combine small partials into scalars ----------------
__global__ void combine_kernel(const float* __restrict__ rowPart,
                               const float* __restrict__ qnumPart,
                               const float* __restrict__ qlogPart,
                               float* __restrict__ acc) {
  if (threadIdx.x == 0) {
    float z = 0.f;
    for (int i = 0; i < 512; ++i) z += rowPart[i];   // sum(P) == 2*sum(Pn), numerically
    float dn = 0.f, lg = 0.f;
    for (int i = 0; i < 32; ++i) { dn += qnumPart[i]; lg += qlogPart[i]; }
    acc[0] = fmaxf(z, EPS_F);                        // Z = max(sum(P), EPS)
    acc[1] = dn;                                     // denom = 2*sum(triu(num))
    acc[2] = lg;                                     // sum_{i!=j} log1p(d_y)
  }
}

// ---------------- K4: nonzero-P entries via directed-edge enumeration ----------------
// Edge e=(i,q): j=nn(i,q). Entry weight w: mutual pair visited twice (w=1), one-sided once (w=2);
// both weightings yield each symmetric nonzero entry exactly twice (entries (i,j) and (j,i)).
__global__ __launch_bounds__(256) void edge_kernel(const float* __restrict__ y,
                                                   const int*   __restrict__ knn_idx,
                                                   const float* __restrict__ knn_p,
                                                   const float* __restrict__ acc,
                                                   float* __restrict__ eT1,
                                                   float* __restrict__ eTQ,
                                                   float* __restrict__ eW) {
  __shared__ float r1[256], r2[256], r3[256];
  const int tid = threadIdx.x;
  const int e   = blockIdx.x * 256 + tid;            // grid 2912 -> exactly 8192*91 edges
  const int i   = e / KNN;
  const int j   = knn_idx[e];
  const float Sij = knn_p[e];

  float Sji = 0.f; bool mutual = false;
  const int* lj = knn_idx + (size_t)j * KNN;
  for (int r = 0; r < KNN; ++r) {
    if (lj[r] == i) { Sji = knn_p[(size_t)j * KNN + r]; mutual = true; break; }
  }

  const float Z = acc[0], denom = acc[1];
  const float2 yi = ((const float2*)y)[i];
  const float2 yj = ((const float2*)y)[j];
  const float dx = yi.x - yj.x, dy = yi.y - yj.y;
  const float d  = dx * dx + dy * dy;
  const float logQ = -log1pf(d) - __logf(denom);

  const float P = (Sij + Sji) / Z;
  const float w = mutual ? 1.0f : 2.0f;
  float t1 = 0.f, tq = 0.f, wn = 0.f;
  if (P > 0.f) {                                     // exp-underflow rows fall to EPS pool
    t1 = w * P * (__logf(P) - logQ);
    tq = w * logQ;
    wn = w;
  }
  r1[tid] = t1; r2[tid] = tq; r3[tid] = wn;
  __syncthreads();
  for (int off = 128; off > 0; off >>= 1) {
    if (tid < off) { r1[tid] += r1[tid + off]; r2[tid] += r2[tid + off]; r3[tid] += r3[tid + off]; }
    __syncthreads();
  }
  if (tid == 0) { eT1[blockIdx.x] = r1[0]; eTQ[blockIdx.x] = r2[0]; eW[blockIdx.x] = r3[0]; }
}

// ---------------- K5: final assembly ----------------
// loss = T1 + EPS * ( Z0*logEPS - (TQ_all - TQnz) )
// TQ_all = sum_all log Q' = -logSum - (N^2-N)*log(denom) + N*logEPS  (diag Q'=EPS)
__global__ __launch_bounds__(256) void final_kernel(const float* __restrict__ eT1,
                                                    const float* __restrict__ eTQ,
                                                    const float* __restrict__ eW,
                                                    const float* __restrict__ acc,
                                                    float* __restrict__ out) {
  __shared__ float r1[256], r2[256], r3[256];
  const int tid = threadIdx.x;
  float a = 0.f, b = 0.f, c = 0.f;
  for (int i = tid; i < 2912; i += 256) { a += eT1[i]; b += eTQ[i]; c += eW[i]; }
  r1[tid] = a; r2[tid] = b; r3[tid] = c;
  __syncthreads();
  for (int off = 128; off > 0; off >>= 1) {
    if (tid < off) { r1[tid] += r1[tid + off]; r2[tid] += r2[tid + off]; r3[tid] += r3[tid + off]; }
    __syncthreads();
  }
  if (tid == 0) {
    const float T1 = r1[0], TQnz = r2[0], nnz = r3[0];
    const float denom = acc[1], logSum = acc[2];
    const float Nf = 8192.0f;
    const float logEps = logf(EPS_F);
    const float tqAll = -logSum - (Nf * Nf - Nf) * __logf(denom) + Nf * logEps;
    const float z0 = Nf * Nf - nnz;
    out[0] = T1 + EPS_F * (z0 * logEps - (tqAll - TQnz));
  }
}

// ---------------- launcher ----------------
extern "C" void kernel_launch(void* const* d_in, const int* in_sizes, int n_in,
                              void* d_out, int out_size, void* d_ws, size_t ws_size,
                              hipStream_t stream) {
  (void)in_sizes; (void)n_in; (void)out_size; (void)ws_size;
  const float* x    = (const float*)d_in[0];
  const float* y    = (const float*)d_in[1];
  const float* beta = (const float*)d_in[2];
  float* out = (float*)d_out;

  char* ws = (char*)d_ws;                            // ~6.04 MB total
  float* sq       = (float*)(ws + 0);                //   32768 B
  int*   knn_idx  = (int*)  (ws + 32768);            // 2981888 B
  float* knn_p    = (float*)(ws + 3014656);          // 2981888 B
  float* rowPart  = (float*)(ws + 5996544);          //    2048 B
  float* qnumPart = (float*)(ws + 5998592);          //     128 B
  float* qlogPart = (float*)(ws + 5998720);          //     128 B
  float* acc      = (float*)(ws + 5998848);          //      16 B
  float* eT1      = (float*)(ws + 5998864);          //   11648 B
  float* eTQ      = (float*)(ws + 6010512);          //   11648 B
  float* eW       = (float*)(ws + 6022160);          //   11648 B

  sq_kernel     <<<NPTS / 8, 256, 0, stream>>>(x, sq);
  knn_kernel    <<<NPTS / 16, 256, 0, stream>>>(x, sq, beta, knn_idx, knn_p, rowPart);
  qsum_kernel   <<<NPTS / 256, 256, 0, stream>>>(y, qnumPart, qlogPart);
  combine_kernel<<<1, 32, 0, stream>>>(rowPart, qnumPart, qlogPart, acc);
  edge_kernel   <<<(NPTS * KNN) / 256, 256, 0, stream>>>(y, knn_idx, knn_p, acc, eT1, eTQ, eW);
  final_kernel  <<<1, 256, 0, stream>>>(eT1, eTQ, eW, acc, out);
}